// SparseNodeAttentionMP_71055938945733
// MI455X (gfx1250) — compile-verified
//
#include <hip/hip_runtime.h>
#include <hip/hip_bf16.h>

// Problem constants (from reference)
constexpr int Bc = 2, Tc = 12, Nc = 10000, Cc = 64, Ec = 160000;
constexpr int BT  = Bc * Tc;          // 24
constexpr int BTN = BT * Nc;          // 240000 rows
constexpr float NEG_SLOPE = 0.2f;

typedef __attribute__((ext_vector_type(2))) float v2f;
typedef __attribute__((ext_vector_type(8))) float v8f;

// ---------------------------------------------------------------------------
// Phase A: feat1/feat2 = x(240000x64) . [a1 | a2]  via V_WMMA_F32_16X16X4_F32
// One wave handles a 16-row tile; 16 WMMA steps cover K=64.
// A (16x4 f32, 2 VGPRs): v0 lanes0-15 K=0 / lanes16-31 K=2; v1: K=1 / K=3.
// B (4x16  f32, 2 VGPRs): same K striping; lane%16 = column. Col0=a1, col1=a2.
// D (16x16 f32, 8 VGPRs): vgpr r, lanes0-15 -> M=r, N=lane; lanes16-31 -> M=8+r.
// B is built branch-free (clamped address + v_cndmask) so EXEC stays all-1s
// and no exec-mask branches serialize the WMMA stream.
// ---------------------------------------------------------------------------
__global__ void __launch_bounds__(256)
feat_wmma_kernel(const float* __restrict__ x,   // [BTN, 64]
                 const float* __restrict__ a,   // [128] (a1=0..63, a2=64..127)
                 float* __restrict__ feat1,     // [BTN]
                 float* __restrict__ feat2) {   // [BTN]
    const int wave = (blockIdx.x * (blockDim.x >> 5)) + (threadIdx.x >> 5); // uniform per wave
    const int lane = threadIdx.x & 31;
    const int row0 = wave * 16;
    if (row0 >= BTN) return;                    // wave-uniform: EXEC stays all-1s

    const int m     = lane & 15;                // row in tile (A) / column (B,D)
    const int khalf = lane >> 4;                // 0 -> K{0,1}, 1 -> K{2,3}
    const float* __restrict__ xrow = x + (size_t)(row0 + m) * Cc;

    // Branch-free B source: column 0 reads a1, column 1 reads a2, columns >=2
    // read a2 too (always in-bounds) but are masked to zero with a select.
    const float colmask = (m < 2) ? 1.0f : 0.0f;
    const float* __restrict__ acol = a + ((m < 2) ? m : 1) * 64;

    v8f acc = {};
    #pragma unroll
    for (int kk = 0; kk < 16; ++kk) {
        const int kb = kk * 4 + khalf * 2;
        v2f A; A.x = xrow[kb]; A.y = xrow[kb + 1];
        v2f Bf; Bf.x = acol[kb] * colmask; Bf.y = acol[kb + 1] * colmask;
        acc = __builtin_amdgcn_wmma_f32_16x16x4_f32(
            /*neg_a=*/false, A, /*neg_b=*/false, Bf,
            /*c_mod=*/(short)0, acc, /*reuse_a=*/false, /*reuse_b=*/false);
    }

    // Column 0 of D = feat1, column 1 = feat2.
    const int mbase = khalf * 8;                // lanes16-31 hold rows 8..15
    if (m == 0) {
        #pragma unroll
        for (int r = 0; r < 8; ++r) feat1[row0 + mbase + r] = acc[r];
    } else if (m == 1) {
        #pragma unroll
        for (int r = 0; r < 8; ++r) feat2[row0 + mbase + r] = acc[r];
    }
}

// ---------------------------------------------------------------------------
// Phase B: denom[bt, dst] += exp(leaky_relu(feat1[bt,dst] + feat2[bt,src]))
// ---------------------------------------------------------------------------
__global__ void __launch_bounds__(256)
edge_denom_kernel(const int* __restrict__ ei,      // [2, E] (dst row, src row)
                  const float* __restrict__ feat1,
                  const float* __restrict__ feat2,
                  float* __restrict__ denom) {     // [BT, N], pre-zeroed
    const long long gid = (long long)blockIdx.x * blockDim.x + threadIdx.x;
    if (gid >= (long long)BT * Ec) return;
    const int e  = (int)(gid % Ec);
    const int bt = (int)(gid / Ec);
    const int d  = ei[e];
    const int s  = ei[Ec + e];
    const size_t bo = (size_t)bt * Nc;
    float v = feat1[bo + d] + feat2[bo + s];
    v = (v > 0.0f) ? v : NEG_SLOPE * v;
    unsafeAtomicAdd(&denom[bo + d], __expf(v));
}

// ---------------------------------------------------------------------------
// Phase C: out[bt,dst,:] += alpha * x[bt,src,:]
// 16 lanes per edge, one float4 per lane -> 256B coalesced gather per edge.
// exp term recomputed (cheap) to avoid a 15MB per-edge spill buffer.
// ---------------------------------------------------------------------------
__global__ void __launch_bounds__(256)
edge_scatter_kernel(const int* __restrict__ ei,
                    const float* __restrict__ x,      // [BT, N, 64]
                    const float* __restrict__ feat1,
                    const float* __restrict__ feat2,
                    const float* __restrict__ denom,  // [BT, N]
                    float* __restrict__ out) {        // [BT, N, 64], pre-zeroed
    const long long gid   = (long long)blockIdx.x * blockDim.x + threadIdx.x;
    const long long group = gid >> 4;                 // one edge per 16 lanes
    const int cl          = (int)(gid & 15);          // channel chunk 0..15
    if (group >= (long long)BT * Ec) return;
    const int e  = (int)(group % Ec);
    const int bt = (int)(group / Ec);
    const int d  = ei[e];
    const int s  = ei[Ec + e];
    const size_t bo = (size_t)bt * Nc;

    float v = feat1[bo + d] + feat2[bo + s];
    v = (v > 0.0f) ? v : NEG_SLOPE * v;
    float den = denom[bo + d];
    den = (den > 1e-8f) ? den : 1e-8f;
    const float alpha = __expf(v) / den;

    const float4 xv = ((const float4*)(x + (bo + s) * Cc))[cl];
    float* op = out + (bo + d) * Cc + cl * 4;
    unsafeAtomicAdd(op + 0, alpha * xv.x);
    unsafeAtomicAdd(op + 1, alpha * xv.y);
    unsafeAtomicAdd(op + 2, alpha * xv.z);
    unsafeAtomicAdd(op + 3, alpha * xv.w);
}

extern "C" void kernel_launch(void* const* d_in, const int* in_sizes, int n_in,
                              void* d_out, int out_size, void* d_ws, size_t ws_size,
                              hipStream_t stream) {
    const float* x  = (const float*)d_in[0];   // [B,T,N,C] f32
    const int*   ei = (const int*)d_in[1];     // [2,E] i32
    const float* a  = (const float*)d_in[2];   // [2C,1] f32
    float* out = (float*)d_out;

    float* feat1 = (float*)d_ws;               // BTN floats
    float* feat2 = feat1 + BTN;                // BTN floats
    float* denom = feat2 + BTN;                // BTN floats

    // Zero accumulation targets (graph-capture-safe async memsets).
    hipMemsetAsync(denom, 0, (size_t)BTN * sizeof(float), stream);
    hipMemsetAsync(out,   0, (size_t)out_size * sizeof(float), stream);

    // Phase A: 240000/16 = 15000 tiles, 8 waves (256 thr) per block.
    {
        const int tiles = BTN / 16;
        const int wpb = 256 / 32;
        feat_wmma_kernel<<<(tiles + wpb - 1) / wpb, 256, 0, stream>>>(x, a, feat1, feat2);
    }
    // Phase B: one thread per (bt, edge).
    {
        const long long total = (long long)BT * Ec;
        edge_denom_kernel<<<(int)((total + 255) / 256), 256, 0, stream>>>(ei, feat1, feat2, denom);
    }
    // Phase C: 16 threads per (bt, edge).
    {
        const long long total = (long long)BT * Ec * 16;
        edge_scatter_kernel<<<(int)((total + 255) / 256), 256, 0, stream>>>(ei, x, feat1, feat2, denom, out);
    }
}